// GCNEncoder_22900765623076
// MI455X (gfx1250) — compile-verified
//
#include <hip/hip_runtime.h>

#define HID 64

typedef float v2f __attribute__((ext_vector_type(2)));
typedef float v8f __attribute__((ext_vector_type(8)));

__device__ __forceinline__ void atomic_add_f32(float* p, float v) {
    // unsafeAtomicAdd is HIP's guaranteed path to the native no-return
    // global_atomic_add_f32 (attaches no-fine-grained-memory metadata so the
    // backend does NOT expand to a global_atomic_cmpswap loop).
    unsafeAtomicAdd(p, v);
}

// ---------------- degree / norm precompute (shared by both layers) ----------
__global__ void deg_init_kernel(float* __restrict__ deg, int N) {
    int i = blockIdx.x * blockDim.x + threadIdx.x;
    if (i < N) deg[i] = 1.0f;  // self-loop weight
}

__global__ void deg_accum_kernel(const int* __restrict__ dst,
                                 const float* __restrict__ ew,
                                 float* __restrict__ deg, int E) {
    int e = blockIdx.x * blockDim.x + threadIdx.x;
    if (e < E) atomic_add_f32(deg + dst[e], ew[e]);
}

__global__ void dinv_kernel(float* __restrict__ deg, int N) {
    int i = blockIdx.x * blockDim.x + threadIdx.x;
    if (i < N) {
        float d = deg[i];
        deg[i] = (d > 0.0f) ? rsqrtf(d) : 0.0f;   // in-place: deg -> dinv
    }
}

__global__ void norm_kernel(const int* __restrict__ src,
                            const int* __restrict__ dst,
                            const float* __restrict__ ew,
                            const float* __restrict__ dinv,
                            float* __restrict__ norm, int E) {
    int e = blockIdx.x * blockDim.x + threadIdx.x;
    if (e < E) norm[e] = dinv[src[e]] * ew[e] * dinv[dst[e]];
}

// ---------------- fp32 WMMA GEMM: H[N,64] = X[N,K] @ W[K,64] ----------------
// One wave computes a 16-row x 64-col slab via V_WMMA_F32_16X16X4_F32.
__global__ __launch_bounds__(128) void gemm_wmma_f32_kernel(
    const float* __restrict__ X, const float* __restrict__ Wm,
    float* __restrict__ H, int N, int K) {
    const int wave = threadIdx.x >> 5;
    const int lane = threadIdx.x & 31;
    const int nRowTiles = (N + 15) >> 4;
    const int rt = blockIdx.x * 4 + wave;
    if (rt >= nRowTiles) return;          // wave-uniform: EXEC stays all-1s

    // A-matrix 16x4 f32 layout: lanes 0-15 rows M=0..15 (K=0,1 in v0,v1),
    // lanes 16-31 same rows (K=2,3).
    int m = (rt << 4) + (lane & 15);
    if (m >= N) m = N - 1;                // clamp (N is a multiple of 16 anyway)
    const int khalf = (lane >> 4) << 1;   // 0 or 2
    const int ncol  = lane & 15;

    v8f acc0 = {}, acc1 = {}, acc2 = {}, acc3 = {};
    const float* xrow = X + (size_t)m * K;

    for (int k = 0; k < K; k += 4) {
        const int ka = k + khalf;
        v2f a;
        a.x = xrow[ka];
        a.y = xrow[ka + 1];
        // B-matrix 4x16 f32: lanes 0-15 hold rows K=ka (v0) / ka+1 (v1) at col=lane,
        // lanes 16-31 hold rows ka(=k+2)/ka+1(=k+3).
        const float* w0 = Wm + (size_t)ka * HID;
        const float* w1 = w0 + HID;
        v2f b0, b1, b2, b3;
        b0.x = w0[ncol];      b0.y = w1[ncol];
        b1.x = w0[16 + ncol]; b1.y = w1[16 + ncol];
        b2.x = w0[32 + ncol]; b2.y = w1[32 + ncol];
        b3.x = w0[48 + ncol]; b3.y = w1[48 + ncol];

        acc0 = __builtin_amdgcn_wmma_f32_16x16x4_f32(false, a, false, b0, (short)0, acc0, false, false);
        acc1 = __builtin_amdgcn_wmma_f32_16x16x4_f32(false, a, false, b1, (short)0, acc1, false, false);
        acc2 = __builtin_amdgcn_wmma_f32_16x16x4_f32(false, a, false, b2, (short)0, acc2, false, false);
        acc3 = __builtin_amdgcn_wmma_f32_16x16x4_f32(false, a, false, b3, (short)0, acc3, false, false);
    }

    // C/D 16x16 f32 layout: VGPR r -> lanes 0-15 row M=r, lanes 16-31 row M=r+8.
    const int mbase = (rt << 4) + ((lane >> 4) << 3);
    #pragma unroll
    for (int r = 0; r < 8; ++r) {
        const int mrow = mbase + r;
        if (mrow < N) {
            float* hr = H + (size_t)mrow * HID + ncol;
            hr[0]  = acc0[r];
            hr[16] = acc1[r];
            hr[32] = acc2[r];
            hr[48] = acc3[r];
        }
    }
}

// ---------------- aggregation: self-loop seed, edge scatter, relu -----------
__global__ void agg_init_kernel(const float* __restrict__ h,
                                const float* __restrict__ dinv,
                                const float* __restrict__ bias,
                                float* __restrict__ agg, int total) {
    int t = blockIdx.x * blockDim.x + threadIdx.x;
    if (t < total) {
        const int n = t >> 6;
        const int d = t & 63;
        const float s = dinv[n];
        agg[t] = h[t] * s * s + bias[d];   // self-loop (norm = dinv^2 * 1) + bias
    }
}

__global__ void scatter_kernel(const int* __restrict__ src,
                               const int* __restrict__ dst,
                               const float* __restrict__ norm,
                               const float* __restrict__ h,
                               float* __restrict__ agg, int total) {
    int t = blockIdx.x * blockDim.x + threadIdx.x;
    if (t >= total) return;
    const int e = t >> 5;                 // edge
    const int d = (t & 31) << 1;          // dim pair
    const int s = src[e];
    const int q = dst[e];
    const float nv = norm[e];
    const float2 v = *(const float2*)(h + (size_t)s * HID + d);  // coalesced, L2-resident
    atomic_add_f32(agg + (size_t)q * HID + d,     v.x * nv);
    atomic_add_f32(agg + (size_t)q * HID + d + 1, v.y * nv);
}

__global__ void relu_kernel(float* __restrict__ p, int total) {
    int t = blockIdx.x * blockDim.x + threadIdx.x;
    if (t < total) p[t] = fmaxf(p[t], 0.0f);
}

// ---------------------------------------------------------------------------
extern "C" void kernel_launch(void* const* d_in, const int* in_sizes, int n_in,
                              void* d_out, int out_size, void* d_ws, size_t ws_size,
                              hipStream_t stream) {
    const float* x    = (const float*)d_in[0];
    const int*   eidx = (const int*)d_in[1];
    const float* ew   = (const float*)d_in[2];
    const float* W1   = (const float*)d_in[3];
    const float* b1   = (const float*)d_in[4];
    const float* W2   = (const float*)d_in[5];
    const float* b2   = (const float*)d_in[6];
    float* out = (float*)d_out;

    const int E = in_sizes[2];            // edge_weight count
    const int N = in_sizes[0] / 128;      // x is [N, 128]
    const int K1 = in_sizes[0] / N;       // 128
    const int* src = eidx;
    const int* dst = eidx + E;

    // workspace layout (floats)
    float* dinv = (float*)d_ws;                              // N (deg -> dinv in place)
    float* norm = dinv + (((size_t)N + 63) & ~(size_t)63);   // E
    float* bufH = norm + (((size_t)E + 63) & ~(size_t)63);   // N*64 (gemm out, reused)
    float* bufA = bufH + (size_t)N * HID;                    // N*64 (layer-1 activations)

    const int TB = 256;
    const int NH = N * HID;
    const int nRowTiles = (N + 15) >> 4;
    const int gemmBlocks = (nRowTiles + 3) / 4;
    const int scatterTotal = E * 32;

    // ---- shared precompute: dinv, per-edge norm ----
    deg_init_kernel<<<(N + TB - 1) / TB, TB, 0, stream>>>(dinv, N);
    deg_accum_kernel<<<(E + TB - 1) / TB, TB, 0, stream>>>(dst, ew, dinv, E);
    dinv_kernel<<<(N + TB - 1) / TB, TB, 0, stream>>>(dinv, N);
    norm_kernel<<<(E + TB - 1) / TB, TB, 0, stream>>>(src, dst, ew, dinv, norm, E);

    // ---- layer 1 ----
    gemm_wmma_f32_kernel<<<gemmBlocks, 128, 0, stream>>>(x, W1, bufH, N, K1);
    agg_init_kernel<<<(NH + TB - 1) / TB, TB, 0, stream>>>(bufH, dinv, b1, bufA, NH);
    scatter_kernel<<<(scatterTotal + TB - 1) / TB, TB, 0, stream>>>(src, dst, norm, bufH, bufA, scatterTotal);
    relu_kernel<<<(NH + TB - 1) / TB, TB, 0, stream>>>(bufA, NH);

    // ---- layer 2 ----
    gemm_wmma_f32_kernel<<<gemmBlocks, 128, 0, stream>>>(bufA, W2, bufH, N, HID);
    agg_init_kernel<<<(NH + TB - 1) / TB, TB, 0, stream>>>(bufH, dinv, b2, out, NH);
    scatter_kernel<<<(scatterTotal + TB - 1) / TB, TB, 0, stream>>>(src, dst, norm, bufH, out, scatterTotal);
    relu_kernel<<<(NH + TB - 1) / TB, TB, 0, stream>>>(out, NH);
}